// SCABlockSparseAdapter_56530359549999
// MI455X (gfx1250) — compile-verified
//
#include <hip/hip_runtime.h>
#include <hip/hip_bf16.h>

// ---- problem constants (match reference) ----
constexpr int NUM_BLOCKS = 64;
constexpr int BLOCK_SIZE = 64;
constexpr int BLOCK_RANK = 16;
constexpr int HIDDEN     = NUM_BLOCKS * BLOCK_SIZE;   // 4096
constexpr int TOP_K      = 4;
constexpr int N_ROWS     = 4 * 2048;                  // 8192
constexpr int MAX_PAIRS  = N_ROWS * TOP_K;            // 32768

typedef __attribute__((ext_vector_type(2))) float v2f;
typedef __attribute__((ext_vector_type(8))) float v8f;

// ---------------------------------------------------------------------------
// Kernel 1: zero the output (128 MB stream) + clear bucket counters
// ---------------------------------------------------------------------------
__global__ __launch_bounds__(256) void zero_kernel(float4* __restrict__ out4,
                                                   size_t n4, int* __restrict__ counts) {
    if (blockIdx.x == 0 && threadIdx.x < NUM_BLOCKS) counts[threadIdx.x] = 0;
    size_t stride = (size_t)gridDim.x * blockDim.x;
    float4 z; z.x = 0.f; z.y = 0.f; z.z = 0.f; z.w = 0.f;
    for (size_t i = (size_t)blockIdx.x * blockDim.x + threadIdx.x; i < n4; i += stride)
        out4[i] = z;
}

// ---------------------------------------------------------------------------
// Kernel 2: per-row dedup + count pairs per block
// ---------------------------------------------------------------------------
__global__ __launch_bounds__(256) void count_kernel(const int* __restrict__ idx,
                                                    int* __restrict__ counts) {
    int r = blockIdx.x * blockDim.x + threadIdx.x;
    if (r >= N_ROWS) return;
    int id[TOP_K];
#pragma unroll
    for (int k = 0; k < TOP_K; ++k) id[k] = idx[r * TOP_K + k];
#pragma unroll
    for (int k = 0; k < TOP_K; ++k) {
        if (id[k] < 0) continue;
        bool first = true;
#pragma unroll
        for (int j = 0; j < TOP_K; ++j)
            if (j < k && id[j] == id[k]) first = false;
        if (first) atomicAdd(&counts[id[k]], 1);
    }
}

// ---------------------------------------------------------------------------
// Kernel 3: tiny exclusive scan over 64 buckets
// ---------------------------------------------------------------------------
__global__ void scan_kernel(const int* __restrict__ counts,
                            int* __restrict__ offs, int* __restrict__ cursors) {
    if (threadIdx.x == 0) {
        int acc = 0;
        for (int i = 0; i < NUM_BLOCKS; ++i) {
            offs[i] = acc; cursors[i] = acc; acc += counts[i];
        }
        offs[NUM_BLOCKS] = acc;
    }
}

// ---------------------------------------------------------------------------
// Kernel 4: softmax + dedup-merge + scatter (row, summed-weight) into buckets
// ---------------------------------------------------------------------------
__global__ __launch_bounds__(256) void scatter_kernel(const int* __restrict__ idx,
                                                      const float* __restrict__ score,
                                                      int* __restrict__ cursors,
                                                      int* __restrict__ pair_row,
                                                      float* __restrict__ pair_sw) {
    int r = blockIdx.x * blockDim.x + threadIdx.x;
    if (r >= N_ROWS) return;
    int id[TOP_K]; float sc[TOP_K]; bool v[TOP_K];
    float mx = -3.4e38f;
#pragma unroll
    for (int k = 0; k < TOP_K; ++k) {
        id[k] = idx[r * TOP_K + k];
        sc[k] = score[r * TOP_K + k];
        v[k]  = (id[k] >= 0);
        if (v[k]) mx = fmaxf(mx, sc[k]);
    }
    float e[TOP_K]; float sum = 0.f;
#pragma unroll
    for (int k = 0; k < TOP_K; ++k) {
        e[k] = v[k] ? __expf(sc[k] - mx) : 0.f;
        sum += e[k];
    }
    float inv = 1.f / sum;
#pragma unroll
    for (int k = 0; k < TOP_K; ++k) {
        if (!v[k]) continue;
        bool first = true;
#pragma unroll
        for (int j = 0; j < TOP_K; ++j)
            if (j < k && id[j] == id[k]) first = false;
        if (!first) continue;
        float s = e[k];
#pragma unroll
        for (int j = 0; j < TOP_K; ++j)
            if (j > k && v[j] && id[j] == id[k]) s += e[j];
        int pos = atomicAdd(&cursors[id[k]], 1);
        pair_row[pos] = r;
        pair_sw[pos]  = s * inv;
    }
}

// ---------------------------------------------------------------------------
// Kernel 5: per-block batched adapter GEMM with f32 WMMA
//   one workgroup per adapter block, 8 waves, 16-row tiles per wave
// ---------------------------------------------------------------------------
#define WMMA_F32(a, b, c) \
    __builtin_amdgcn_wmma_f32_16x16x4_f32(false, (a), false, (b), (short)0, (c), false, false)

__global__ __launch_bounds__(256) void gemm_kernel(const float* __restrict__ hidden,
                                                   const float* __restrict__ down_w,
                                                   const float* __restrict__ down_b,
                                                   const float* __restrict__ up_w,
                                                   const float* __restrict__ up_b,
                                                   const int* __restrict__ offs,
                                                   const int* __restrict__ pair_row,
                                                   const float* __restrict__ pair_sw,
                                                   float* __restrict__ out) {
    constexpr int XS = 68;   // X tile LDS stride  (bank-conflict-free A reads)
    constexpr int RS = 20;   // rank tile LDS stride
    constexpr int US = 72;   // W_up LDS stride    (half-waves hit disjoint banks)

    __shared__ __align__(16) float wd_s[BLOCK_SIZE * BLOCK_RANK];  // [64][16]
    __shared__ __align__(16) float wu_s[BLOCK_RANK * US];          // [16][72]
    __shared__ __align__(16) float xt_s[8][16 * XS];
    __shared__ __align__(16) float rk_s[8][16 * RS];
    __shared__ int   rows_s[8][16];
    __shared__ float sw_s[8][16];

    const int b    = blockIdx.x;
    const int lane = threadIdx.x & 31;
    const int wv   = threadIdx.x >> 5;
    const int half = lane >> 4;      // 0: lanes 0-15, 1: lanes 16-31
    const int n    = lane & 15;      // N index / A-row index

    // stage this block's adapter weights into LDS
    for (int i = threadIdx.x; i < BLOCK_SIZE * BLOCK_RANK; i += blockDim.x) {
        wd_s[i] = down_w[b * BLOCK_SIZE * BLOCK_RANK + i];         // [64][16] row-major
        int k = i >> 6, j = i & 63;                                // up_w: [16][64]
        wu_s[k * US + j] = up_w[b * BLOCK_RANK * BLOCK_SIZE + i];
    }
    __syncthreads();

    const int base0 = offs[b];
    const int cnt   = offs[b + 1] - base0;
    const int ntile = (cnt + 15) >> 4;

    const float bdn = down_b[b * BLOCK_RANK + n];
    float* xt = xt_s[wv];
    float* rk = rk_s[wv];

    for (int t = wv; t < ntile; t += 8) {
        const int base  = base0 + t * 16;
        const int valid = min(16, cnt - t * 16);

        // fetch this tile's row ids + softmax weights
        if (lane < 16) {
            int row = 0; float sw = 0.f;
            if (lane < valid) { row = pair_row[base + lane]; sw = pair_sw[base + lane]; }
            rows_s[wv][lane] = row;
            sw_s[wv][lane]   = sw;
        }
        __builtin_amdgcn_wave_barrier();

        // gather X tile [16 rows x 64] (16B vectors, coalesced per row segment)
        for (int j = lane; j < 16 * 16; j += 32) {
            int m = j >> 4, q = j & 15;
            float4 v; v.x = 0.f; v.y = 0.f; v.z = 0.f; v.w = 0.f;
            if (m < valid) {
                const float4* src = (const float4*)(hidden +
                    (size_t)rows_s[wv][m] * HIDDEN + b * BLOCK_SIZE + q * 4);
                v = *src;
            }
            *(float4*)(&xt[m * XS + q * 4]) = v;   // (m*68+q*4)*4B is 16B aligned
        }
        __builtin_amdgcn_wave_barrier();

        // GEMM #1: rank[16x16] = X[16x64] @ W_down[64x16]  (16 chained K=4 WMMAs)
        v8f acc = {};
#pragma unroll
        for (int s = 0; s < 16; ++s) {
            const int k0 = 4 * s + 2 * half;
            v2f a, bm;
            a.x  = xt[n * XS + k0];
            a.y  = xt[n * XS + k0 + 1];
            bm.x = wd_s[k0 * BLOCK_RANK + n];
            bm.y = wd_s[(k0 + 1) * BLOCK_RANK + n];
            acc = WMMA_F32(a, bm, acc);
        }

        // bias + silu, relayout C->A through LDS
#pragma unroll
        for (int r = 0; r < 8; ++r) {
            int m = r + 8 * half;
            float v = acc[r] + bdn;
            v = v * (1.f / (1.f + __expf(-v)));
            rk[m * RS + n] = v;
        }
        __builtin_amdgcn_wave_barrier();

        // GEMM #2: out[16x64] = rank[16x16] @ W_up[16x64]  (4 N-tiles x 4 K-steps)
#pragma unroll
        for (int nt = 0; nt < 4; ++nt) {
            v8f acc2 = {};
#pragma unroll
            for (int s = 0; s < 4; ++s) {
                const int k0 = 4 * s + 2 * half;
                v2f a, bm;
                a.x  = rk[n * RS + k0];
                a.y  = rk[n * RS + k0 + 1];
                bm.x = wu_s[k0 * US + nt * 16 + n];
                bm.y = wu_s[(k0 + 1) * US + nt * 16 + n];
                acc2 = WMMA_F32(a, bm, acc2);
            }
            const float bup = up_b[b * BLOCK_SIZE + nt * 16 + n];
#pragma unroll
            for (int r = 0; r < 8; ++r) {
                int m = r + 8 * half;
                if (m < valid) {
                    float val = (acc2[r] + bup) * sw_s[wv][m];
                    // dedup guarantees each (row, block) written exactly once
                    out[(size_t)rows_s[wv][m] * HIDDEN + b * BLOCK_SIZE + nt * 16 + n] = val;
                }
            }
        }
        __builtin_amdgcn_wave_barrier();
    }
}

// ---------------------------------------------------------------------------
extern "C" void kernel_launch(void* const* d_in, const int* in_sizes, int n_in,
                              void* d_out, int out_size, void* d_ws, size_t ws_size,
                              hipStream_t stream) {
    const float* hidden = (const float*)d_in[0];
    const int*   aidx   = (const int*)d_in[1];
    const float* ascore = (const float*)d_in[2];
    const float* down_w = (const float*)d_in[3];
    const float* down_b = (const float*)d_in[4];
    const float* up_w   = (const float*)d_in[5];
    const float* up_b   = (const float*)d_in[6];
    float*       out    = (float*)d_out;

    int*   wsi     = (int*)d_ws;
    int*   counts  = wsi;                       // 64
    int*   offs    = wsi + 64;                  // 65
    int*   cursors = wsi + 192;                 // 64
    int*   pairs   = wsi + 256;                 // 32768
    float* psw     = (float*)(wsi + 256 + MAX_PAIRS);  // 32768

    const size_t n4 = (size_t)N_ROWS * HIDDEN / 4;
    zero_kernel<<<2048, 256, 0, stream>>>((float4*)out, n4, counts);
    count_kernel<<<N_ROWS / 256, 256, 0, stream>>>(aidx, counts);
    scan_kernel<<<1, 64, 0, stream>>>(counts, offs, cursors);
    scatter_kernel<<<N_ROWS / 256, 256, 0, stream>>>(aidx, ascore, cursors, pairs, psw);
    gemm_kernel<<<NUM_BLOCKS, 256, 0, stream>>>(hidden, down_w, down_b, up_w, up_b,
                                                offs, pairs, psw, out);
}